// NovaIDEIA_14843406975526
// MI455X (gfx1250) — compile-verified
//
#include <hip/hip_runtime.h>
#include <hip/hip_bf16.h>
#include <math.h>

// ---------------------------------------------------------------------------
// VQ-VAE 3D forward for MI455X (gfx1250, wave32, WMMA).
// Convolutions run as implicit GEMM on v_wmma_f32_16x16x4_f32:
//   D(16x16) += A(16x4) x B(4x16); M = 16 output w-positions, N = 16 out chans
// Transposed convs are parity-decomposed (2 valid taps/dim) so K_eff = 8*Cin.
// Interior tiles use a precomputed flat-offset table (no bounds checks).
// ---------------------------------------------------------------------------

typedef float v2f __attribute__((ext_vector_type(2)));
typedef float v8f __attribute__((ext_vector_type(8)));

#define WAVES_PER_BLOCK 8
#define TPB (WAVES_PER_BLOCK * 32)

__device__ __forceinline__ float act_apply(float v, int ACT) {
    if (ACT == 1) v = fmaxf(v, 0.0f);
    if (ACT == 2) v = fminf(fmaxf(v, 0.0f), 6.0f);
    return v;
}

// ---------------------------------------------------------------------------
// Weight packing: B panel [n(16)][K], zero-padded for n >= COUT.
// ---------------------------------------------------------------------------
template <int CIN, int COUT, int KS, bool FLIP>
__global__ void pack_conv_b(const float* __restrict__ w, float* __restrict__ Bp) {
    constexpr int K = CIN * KS * KS * KS;
    int i = blockIdx.x * blockDim.x + threadIdx.x;
    if (i >= 16 * K) return;
    int nn = i / K, k = i % K;
    float v = 0.0f;
    if (nn < COUT) {
        int ci = k / (KS * KS * KS), r = k % (KS * KS * KS);
        int kd = r / (KS * KS), kh = (r / KS) % KS, kw = r % KS;
        if (FLIP) {
            // deconv stride-1 == conv with flipped kernel; W layout (CIN,COUT,k,k,k)
            v = w[(((ci * COUT + nn) * KS + (KS - 1 - kd)) * KS + (KS - 1 - kh)) * KS + (KS - 1 - kw)];
        } else {
            // W layout (COUT,CIN,k,k,k)
            v = w[(((nn * CIN + ci) * KS + kd) * KS + kh) * KS + kw];
        }
    }
    Bp[i] = v;
}

// Transposed conv (k=4, s=2, p=1): pack per output-parity class q = qd*4+qh*2+qw.
template <int CIN, int COUT>
__global__ void pack_deconv_b(const float* __restrict__ w, float* __restrict__ Bp) {
    constexpr int K = CIN * 8;
    int i = blockIdx.x * blockDim.x + threadIdx.x;
    if (i >= 8 * 16 * K) return;
    int q = i / (16 * K), rr = i % (16 * K);
    int nn = rr / K, k = rr % K;
    int qd = (q >> 2) & 1, qh = (q >> 1) & 1, qw = q & 1;
    float v = 0.0f;
    if (nn < COUT) {
        int ci = k >> 3, t = k & 7;
        int kd = qd + 2 * ((t >> 2) & 1);
        int kh = qh + 2 * ((t >> 1) & 1);
        int kw = qw + 2 * (t & 1);
        v = w[(((ci * COUT + nn) * 4 + kd) * 4 + kh) * 4 + kw];   // (CIN,COUT,4,4,4)
    }
    Bp[i] = v;
}

// ---------------------------------------------------------------------------
// Direct conv implicit-GEMM. One wave per M-tile (16 consecutive ow).
// ---------------------------------------------------------------------------
template <int CIN, int COUT, int KS, int STRIDE, int PAD, int ACT>
__global__ __launch_bounds__(TPB) void conv_wmma(
    const float* __restrict__ in, const float* __restrict__ Bp,
    const float* __restrict__ bias, float* __restrict__ out,
    int N, int ID, int IH, int IW, int OD, int OH, int OW) {
    constexpr int K = CIN * KS * KS * KS;
    constexpr int KSTEPS = K / 4;
    __shared__ __align__(8) int otab[K];       // flat input offset per k
    __shared__ unsigned ktab[K];               // packed (ci,kd,kh,kw) for border path
    for (int k = threadIdx.x; k < K; k += blockDim.x) {
        int ci = k / (KS * KS * KS), r = k % (KS * KS * KS);
        int kd = r / (KS * KS), kh = (r / KS) % KS, kw = r % KS;
        otab[k] = ((ci * ID + kd) * IH + kh) * IW + kw;
        ktab[k] = ((unsigned)ci << 24) | ((unsigned)kd << 16) | ((unsigned)kh << 8) | (unsigned)kw;
    }
    __syncthreads();

    const int lane = threadIdx.x & 31;
    const int wave = threadIdx.x >> 5;
    const int tiles_w = OW >> 4;
    const long total = (long)N * OD * OH * tiles_w;
    long tile = (long)blockIdx.x * WAVES_PER_BLOCK + wave;
    if (tile >= total) return;
    int owt = (int)(tile % tiles_w); long t1 = tile / tiles_w;
    int oh = (int)(t1 % OH); t1 /= OH;
    int od = (int)(t1 % OD); int n = (int)(t1 / OD);

    const int mA = lane & 15;       // A-matrix row (output position)
    const int kp = lane >> 4;       // K-pair selector
    const int ow = (owt << 4) + mA;
    const int idb = od * STRIDE - PAD;
    const int ihb = oh * STRIDE - PAD;
    const int iwb = ow * STRIDE - PAD;
    const int iwb0 = (owt << 4) * STRIDE - PAD;
    const float* inb = in + (size_t)n * CIN * ID * IH * IW;
    const float* Brow = Bp + (size_t)(lane & 15) * K;   // B-matrix column n

    const bool interior = (idb >= 0) & (idb + KS <= ID) & (ihb >= 0) & (ihb + KS <= IH) &
                          (iwb0 >= 0) & (iwb0 + 15 * STRIDE + KS <= IW);

    v8f acc = {0.f, 0.f, 0.f, 0.f, 0.f, 0.f, 0.f, 0.f};
    if (interior) {
        const float* pb = inb + (size_t)((idb * IH + ihb) * IW + iwb);
#pragma unroll 4
        for (int ks = 0; ks < KSTEPS; ++ks) {
            const int k0 = ks * 4 + kp * 2;
            int2 offs = *(const int2*)&otab[k0];
            v2f a, b;
            a.x = pb[offs.x];
            a.y = pb[offs.y];
            b = *(const v2f*)(Brow + k0);
            acc = __builtin_amdgcn_wmma_f32_16x16x4_f32(false, a, false, b, (short)0, acc,
                                                        false, false);
        }
    } else {
#pragma unroll 2
        for (int ks = 0; ks < KSTEPS; ++ks) {
            const int k0 = ks * 4 + kp * 2;
            unsigned p0 = ktab[k0], p1 = ktab[k0 + 1];
            v2f a, b;
            {
                int ci = p0 >> 24, kd = (p0 >> 16) & 255, kh = (p0 >> 8) & 255, kw = p0 & 255;
                int id = idb + kd, ih = ihb + kh, iw = iwb + kw;
                bool ok = ((unsigned)id < (unsigned)ID) & ((unsigned)ih < (unsigned)IH) &
                          ((unsigned)iw < (unsigned)IW);
                a.x = ok ? inb[(((size_t)ci * ID + id) * IH + ih) * IW + iw] : 0.0f;
            }
            {
                int ci = p1 >> 24, kd = (p1 >> 16) & 255, kh = (p1 >> 8) & 255, kw = p1 & 255;
                int id = idb + kd, ih = ihb + kh, iw = iwb + kw;
                bool ok = ((unsigned)id < (unsigned)ID) & ((unsigned)ih < (unsigned)IH) &
                          ((unsigned)iw < (unsigned)IW);
                a.y = ok ? inb[(((size_t)ci * ID + id) * IH + ih) * IW + iw] : 0.0f;
            }
            b = *(const v2f*)(Brow + k0);
            acc = __builtin_amdgcn_wmma_f32_16x16x4_f32(false, a, false, b, (short)0, acc,
                                                        false, false);
        }
    }

    const int co = lane & 15;
    if (co < COUT) {
        const float bv = bias[co];
        const size_t obase = (((size_t)n * COUT + co) * OD + od) * (size_t)OH * OW + (size_t)oh * OW;
#pragma unroll
        for (int r = 0; r < 8; ++r) {
            int m_o = r + 8 * kp;                    // D: vgpr r -> M=r (lanes<16) / r+8
            out[obase + (owt << 4) + m_o] = act_apply(acc[r] + bv, ACT);
        }
    }
}

// ---------------------------------------------------------------------------
// Transposed conv (k=4, s=2, p=1), parity-decomposed implicit GEMM.
// M-tile = 16 outputs with stride 2 in w (uniform w-parity per wave).
// ---------------------------------------------------------------------------
template <int CIN, int COUT, int ACT>
__global__ __launch_bounds__(TPB) void deconv_s2_wmma(
    const float* __restrict__ in, const float* __restrict__ Bp,
    const float* __restrict__ bias, float* __restrict__ out,
    int N, int ID, int IH, int IW, int OD, int OH, int OW) {
    constexpr int K = CIN * 8;
    constexpr int KSTEPS = K / 4;
    __shared__ __align__(8) int otab[K];
    __shared__ unsigned ktab[K];
    for (int k = threadIdx.x; k < K; k += blockDim.x) {
        int ci = k >> 3, t = k & 7;
        int td = (t >> 2) & 1, th = (t >> 1) & 1, tw = t & 1;
        otab[k] = ci * ID * IH * IW - (td * IH * IW + th * IW + tw);
        ktab[k] = ((unsigned)ci << 24) | ((unsigned)td << 16) | ((unsigned)th << 8) | (unsigned)tw;
    }
    __syncthreads();

    const int lane = threadIdx.x & 31;
    const int wave = threadIdx.x >> 5;
    const int tiles_w = OW >> 5;   // 32 outputs per tile span (16 of one parity)
    const long total = (long)N * OD * OH * tiles_w * 2;
    long tile = (long)blockIdx.x * WAVES_PER_BLOCK + wave;
    if (tile >= total) return;
    int s = (int)(tile & 1); long t1 = tile >> 1;
    int owt = (int)(t1 % tiles_w); t1 /= tiles_w;
    int oh = (int)(t1 % OH); t1 /= OH;
    int od = (int)(t1 % OD); int n = (int)(t1 / OD);

    const int mA = lane & 15;
    const int kp = lane >> 4;
    const int ow0 = (owt << 5) + s;
    const int ow = ow0 + 2 * mA;
    const int qd = (od + 1) & 1, qh = (oh + 1) & 1, qw = (ow0 + 1) & 1;
    const int q = qd * 4 + qh * 2 + qw;
    const int idb = (od + 1 - qd) >> 1;   // id = idb - td
    const int ihb = (oh + 1 - qh) >> 1;
    const int iwb0 = (ow0 + 1 - qw) >> 1;
    const int iwb = iwb0 + mA;
    const float* inb = in + (size_t)n * CIN * ID * IH * IW;
    const float* Brow = Bp + ((size_t)q * 16 + (lane & 15)) * K;

    const bool interior = (idb >= 1) & (idb < ID) & (ihb >= 1) & (ihb < IH) &
                          (iwb0 >= 1) & (iwb0 + 15 < IW);

    v8f acc = {0.f, 0.f, 0.f, 0.f, 0.f, 0.f, 0.f, 0.f};
    if (interior) {
        const float* pb = inb + (size_t)((idb * IH + ihb) * IW + iwb);
#pragma unroll 4
        for (int ks = 0; ks < KSTEPS; ++ks) {
            const int k0 = ks * 4 + kp * 2;
            int2 offs = *(const int2*)&otab[k0];
            v2f a, b;
            a.x = pb[offs.x];
            a.y = pb[offs.y];
            b = *(const v2f*)(Brow + k0);
            acc = __builtin_amdgcn_wmma_f32_16x16x4_f32(false, a, false, b, (short)0, acc,
                                                        false, false);
        }
    } else {
#pragma unroll 2
        for (int ks = 0; ks < KSTEPS; ++ks) {
            const int k0 = ks * 4 + kp * 2;
            unsigned p0 = ktab[k0], p1 = ktab[k0 + 1];
            v2f a, b;
            {
                int ci = p0 >> 24, td = (p0 >> 16) & 255, th = (p0 >> 8) & 255, tw = p0 & 255;
                int id = idb - td, ih = ihb - th, iw = iwb - tw;
                bool ok = ((unsigned)id < (unsigned)ID) & ((unsigned)ih < (unsigned)IH) &
                          ((unsigned)iw < (unsigned)IW);
                a.x = ok ? inb[(((size_t)ci * ID + id) * IH + ih) * IW + iw] : 0.0f;
            }
            {
                int ci = p1 >> 24, td = (p1 >> 16) & 255, th = (p1 >> 8) & 255, tw = p1 & 255;
                int id = idb - td, ih = ihb - th, iw = iwb - tw;
                bool ok = ((unsigned)id < (unsigned)ID) & ((unsigned)ih < (unsigned)IH) &
                          ((unsigned)iw < (unsigned)IW);
                a.y = ok ? inb[(((size_t)ci * ID + id) * IH + ih) * IW + iw] : 0.0f;
            }
            b = *(const v2f*)(Brow + k0);
            acc = __builtin_amdgcn_wmma_f32_16x16x4_f32(false, a, false, b, (short)0, acc,
                                                        false, false);
        }
    }

    const int co = lane & 15;
    if (co < COUT) {
        const float bv = bias[co];
        const size_t obase = (((size_t)n * COUT + co) * OD + od) * (size_t)OH * OW + (size_t)oh * OW;
#pragma unroll
        for (int r = 0; r < 8; ++r) {
            int m_o = r + 8 * kp;
            out[obase + ow0 + 2 * m_o] = act_apply(acc[r] + bv, ACT);
        }
    }
}

// ---------------------------------------------------------------------------
// VQ nearest-neighbor quantization + histogram + loss accumulation.
// z layout NCDHW (16,16,4,32,32); 65536 positions, 32 codes of dim 16.
// ---------------------------------------------------------------------------
#define NPOS 65536
#define SPAT 4096

__global__ __launch_bounds__(256) void vq_kernel(
    const float* __restrict__ z, const float* __restrict__ emb,
    float* __restrict__ quant, float* __restrict__ codes_out,
    float* __restrict__ vqsum, unsigned* __restrict__ hist) {
    __shared__ float se[32 * 16];
    __shared__ unsigned shist[32];
    __shared__ float ssum;
    int t = threadIdx.x;
    if (t < 512) se[t] = emb[t];
    if (t < 32) shist[t] = 0u;
    if (t == 0) ssum = 0.0f;
    __syncthreads();

    int p = blockIdx.x * 256 + t;
    int n = p >> 12, sp = p & (SPAT - 1);
    const float* zb = z + (size_t)n * 16 * SPAT + sp;
    float zv[16];
#pragma unroll
    for (int c = 0; c < 16; ++c) zv[c] = zb[(size_t)c * SPAT];

    float best = 3.4e38f;
    int bidx = 0;
#pragma unroll 4
    for (int j = 0; j < 32; ++j) {
        float d = 0.0f;
#pragma unroll
        for (int c = 0; c < 16; ++c) {
            float df = zv[c] - se[j * 16 + c];
            d = fmaf(df, df, d);
        }
        if (d < best) { best = d; bidx = j; }
    }

    float* qb = quant + (size_t)n * 16 * SPAT + sp;
#pragma unroll
    for (int c = 0; c < 16; ++c) qb[(size_t)c * SPAT] = se[bidx * 16 + c];
    codes_out[p] = (float)bidx;

    atomicAdd(&shist[bidx], 1u);
    atomicAdd(&ssum, best);
    __syncthreads();
    if (t < 32) atomicAdd(&hist[t], shist[t]);
    if (t == 0) atomicAdd(vqsum, ssum);
}

__global__ void finalize_kernel(const unsigned* __restrict__ hist,
                                const float* __restrict__ vqsum,
                                const float* __restrict__ cluster_size,
                                float* __restrict__ o_loss,
                                float* __restrict__ o_perp,
                                float* __restrict__ o_used) {
    if (threadIdx.x != 0 || blockIdx.x != 0) return;
    float H = 0.0f;
    for (int j = 0; j < 32; ++j) {
        float pj = (float)hist[j] / (float)NPOS;
        H += pj * logf(pj + 1e-10f);
    }
    *o_perp = expf(-H);
    int used = 0;
    for (int j = 0; j < 32; ++j) used += (cluster_size[j] > 1e-5f) ? 1 : 0;
    *o_used = (float)used / 32.0f;
    *o_loss = 0.25f * vqsum[0] / ((float)NPOS * 16.0f);
}

// ---------------------------------------------------------------------------
// Host-side orchestration.
// ---------------------------------------------------------------------------
static inline int grid_for(long tiles) { return (int)((tiles + WAVES_PER_BLOCK - 1) / WAVES_PER_BLOCK); }

extern "C" void kernel_launch(void* const* d_in, const int* in_sizes, int n_in,
                              void* d_out, int out_size, void* d_ws, size_t ws_size,
                              hipStream_t stream) {
    const float* x      = (const float*)d_in[0];
    const float* enc_w1 = (const float*)d_in[2];
    const float* enc_b1 = (const float*)d_in[3];
    const float* enc_w2 = (const float*)d_in[4];
    const float* enc_b2 = (const float*)d_in[5];
    const float* enc_w4 = (const float*)d_in[6];
    const float* enc_b4 = (const float*)d_in[7];
    const float* dec_w1 = (const float*)d_in[8];
    const float* dec_b1 = (const float*)d_in[9];
    const float* dec_w3 = (const float*)d_in[10];
    const float* dec_b3 = (const float*)d_in[11];
    const float* dec_w4 = (const float*)d_in[12];
    const float* dec_b4 = (const float*)d_in[13];
    const float* emb    = (const float*)d_in[14];
    const float* clsz   = (const float*)d_in[15];

    float* OUT = (float*)d_out;
    const size_t OUT_ELEMS = 25165824;           // (16,3,16,128,128)
    float* o_loss  = OUT + OUT_ELEMS;
    float* o_codes = OUT + OUT_ELEMS + 1;
    float* o_perp  = OUT + OUT_ELEMS + 1 + NPOS;
    float* o_used  = OUT + OUT_ELEMS + 1 + NPOS + 1;

    // workspace layout (floats)
    float* W = (float*)d_ws;
    size_t o = 0;
    float* z     = W + o; o += 1048576;          // (16,16,4,32,32)
    float* quant = W + o; o += 1048576;
    float* h1    = W + o; o += 2097152;          // (16,4,8,64,64)
    float* h2    = W + o; o += 524288;           // (16,8,4,32,32)
    float* d1    = W + o; o += 4194304;          // (16,8,8,64,64)
    float* d2    = W + o; o += 16777216;         // (16,4,16,128,128)
    float* Bp_e1 = W + o; o += 16 * 192;
    float* Bp_e2 = W + o; o += 16 * 256;
    float* Bp_e3 = W + o; o += 16 * 216;
    float* Bp_d1 = W + o; o += 8 * 16 * 128;
    float* Bp_d2 = W + o; o += 8 * 16 * 64;
    float* Bp_d3 = W + o; o += 16 * 108;
    float* vqsum = W + o; o += 1;
    unsigned* hist = (unsigned*)(W + o); o += 32;

    // zero accumulators (vqsum + hist are contiguous)
    hipMemsetAsync(vqsum, 0, 33 * sizeof(float), stream);

    // pack weight panels
    pack_conv_b<3, 4, 4, false><<<(16 * 192 + 255) / 256, 256, 0, stream>>>(enc_w1, Bp_e1);
    pack_conv_b<4, 8, 4, false><<<(16 * 256 + 255) / 256, 256, 0, stream>>>(enc_w2, Bp_e2);
    pack_conv_b<8, 16, 3, false><<<(16 * 216 + 255) / 256, 256, 0, stream>>>(enc_w4, Bp_e3);
    pack_deconv_b<16, 8><<<(8 * 16 * 128 + 255) / 256, 256, 0, stream>>>(dec_w1, Bp_d1);
    pack_deconv_b<8, 4><<<(8 * 16 * 64 + 255) / 256, 256, 0, stream>>>(dec_w3, Bp_d2);
    pack_conv_b<4, 3, 3, true><<<(16 * 108 + 255) / 256, 256, 0, stream>>>(dec_w4, Bp_d3);

    // encoder
    conv_wmma<3, 4, 4, 2, 1, 1><<<grid_for(16L * 8 * 64 * 4), TPB, 0, stream>>>(
        x, Bp_e1, enc_b1, h1, 16, 16, 128, 128, 8, 64, 64);
    conv_wmma<4, 8, 4, 2, 1, 1><<<grid_for(16L * 4 * 32 * 2), TPB, 0, stream>>>(
        h1, Bp_e2, enc_b2, h2, 16, 8, 64, 64, 4, 32, 32);
    conv_wmma<8, 16, 3, 1, 1, 2><<<grid_for(16L * 4 * 32 * 2), TPB, 0, stream>>>(
        h2, Bp_e3, enc_b4, z, 16, 4, 32, 32, 4, 32, 32);

    // vector quantization
    vq_kernel<<<NPOS / 256, 256, 0, stream>>>(z, emb, quant, o_codes, vqsum, hist);

    // decoder
    deconv_s2_wmma<16, 8, 1><<<grid_for(16L * 8 * 64 * 2 * 2), TPB, 0, stream>>>(
        quant, Bp_d1, dec_b1, d1, 16, 4, 32, 32, 8, 64, 64);
    deconv_s2_wmma<8, 4, 1><<<grid_for(16L * 16 * 128 * 4 * 2), TPB, 0, stream>>>(
        d1, Bp_d2, dec_b3, d2, 16, 8, 64, 64, 16, 128, 128);
    conv_wmma<4, 3, 3, 1, 1, 0><<<grid_for(16L * 16 * 128 * 8), TPB, 0, stream>>>(
        d2, Bp_d3, dec_b4, OUT, 16, 16, 128, 128, 16, 128, 128);

    // scalar outputs
    finalize_kernel<<<1, 32, 0, stream>>>(hist, vqsum, clsz, o_loss, o_perp, o_used);

    (void)in_sizes; (void)n_in; (void)out_size; (void)ws_size;
}